// AdaptiveMemory_28355374088932
// MI455X (gfx1250) — compile-verified
//
#include <hip/hip_runtime.h>
#include <hip/hip_bf16.h>
#include <math.h>

typedef __attribute__((ext_vector_type(2))) float v2f;
typedef __attribute__((ext_vector_type(8))) float v8f;

#define D_DIM 1024
#define ROWS_PER_BLOCK 16
#define THREADS 512                 // 16 wave32 waves
#define WAVES (THREADS / 32)
#define K_SLICE (D_DIM / WAVES)     // 64 K per wave
#define WMMA_ITERS (K_SLICE / 4)    // 16 x V_WMMA_F32_16X16X4_F32 per wave

__global__ __launch_bounds__(THREADS)
void AdaptiveMemory_wmma_kernel(const float* __restrict__ h_last,
                                const int*   __restrict__ vids,
                                const float* __restrict__ mem,
                                const float* __restrict__ W_alpha,
                                const float* __restrict__ b_alpha,
                                const float* __restrict__ decay_p,
                                float* __restrict__ out,
                                int B)
{
    __shared__ float dots[ROWS_PER_BLOCK];

    const int t        = (int)threadIdx.x;
    const int lane     = t & 31;          // wave32 lane id (blockDim multiple of 32)
    const int wave     = t >> 5;
    const int row_base = (int)blockIdx.x * ROWS_PER_BLOCK;

    if (t < ROWS_PER_BLOCK) dots[t] = 0.0f;
    __syncthreads();

    // ---------------- Phase A: alpha dot-products via V_WMMA_F32_16X16X4_F32 ----
    // A (16x4 f32) layout: lane l -> M = l&15 ; VGPR0 = K {0 | 2}, VGPR1 = K {1 | 3}
    // => lane l supplies float2 h[M][k0 + 2*(l>>4)].
    // B (4x16) has every column equal to w, so lane l supplies the same float2 of W_alpha.
    {
        const int m    = lane & 15;
        const int koff = (lane >> 4) * 2;
        const float* hrow = h_last + (size_t)(row_base + m) * D_DIM;

        v8f acc = {};
        int k = wave * K_SLICE + koff;
        #pragma unroll
        for (int j = 0; j < WMMA_ITERS; ++j) {
            v2f a = *reinterpret_cast<const v2f*>(hrow + k);       // 8B aligned (k even)
            v2f b = *reinterpret_cast<const v2f*>(W_alpha + k);    // broadcast across lanes 0-15
            acc = __builtin_amdgcn_wmma_f32_16x16x4_f32(
                      /*neg_a=*/false, a, /*neg_b=*/false, b,
                      /*c_mod=*/(short)0, acc,
                      /*reuse_a=*/false, /*reuse_b=*/false);
            k += 4;
        }

        // D column 0 holds the row dots: lane 0 -> rows 0..7 (v0..v7),
        // lane 16 -> rows 8..15. Reduce the 16 waves' partials in LDS.
        if (lane == 0) {
            #pragma unroll
            for (int r = 0; r < 8; ++r) atomicAdd(&dots[r], acc[r]);
        } else if (lane == 16) {
            #pragma unroll
            for (int r = 0; r < 8; ++r) atomicAdd(&dots[8 + r], acc[r]);
        }
    }
    __syncthreads();

    // ---------------- Phase B: gated elementwise update (coalesced float4) ------
    const int r   = t >> 5;               // 32 threads per row
    const int g   = t & 31;
    const int row = row_base + r;
    if (row >= B) return;

    const float dv    = decay_p[0];
    const float x     = dots[r] + b_alpha[0];
    const float alpha = 1.0f / (1.0f + __expf(-x));
    // M_smoothed = M*(d + (1-d)*alpha) + h*(1-d)*(1-alpha)
    const float cM = dv + (1.0f - dv) * alpha;
    const float cH = (1.0f - dv) * (1.0f - alpha);

    const float4* hr = reinterpret_cast<const float4*>(h_last + (size_t)row * D_DIM);
    const float4* mr = reinterpret_cast<const float4*>(mem + (size_t)vids[row] * D_DIM);
    float4*       orow = reinterpret_cast<float4*>(out + (size_t)row * D_DIM);

    #pragma unroll
    for (int i = 0; i < D_DIM / (32 * 4); ++i) {   // 8 iterations of 256B per half-wave
        const int c4 = g + i * 32;
        const float4 hv = hr[c4];
        const float4 mv = mr[c4];
        float4 o;
        o.x = cM * mv.x + cH * hv.x;
        o.y = cM * mv.y + cH * hv.y;
        o.z = cM * mv.z + cH * hv.z;
        o.w = cM * mv.w + cH * hv.w;
        orow[c4] = o;
    }
}

extern "C" void kernel_launch(void* const* d_in, const int* in_sizes, int n_in,
                              void* d_out, int out_size, void* d_ws, size_t ws_size,
                              hipStream_t stream) {
    (void)n_in; (void)out_size; (void)d_ws; (void)ws_size;
    const float* h_last  = (const float*)d_in[0];
    const int*   vids    = (const int*)  d_in[1];
    const float* mem     = (const float*)d_in[2];
    const float* W_alpha = (const float*)d_in[3];
    const float* b_alpha = (const float*)d_in[4];
    const float* decay   = (const float*)d_in[5];
    float*       out     = (float*)d_out;

    const int B = in_sizes[0] / D_DIM;                     // 4096
    const int grid = (B + ROWS_PER_BLOCK - 1) / ROWS_PER_BLOCK;  // 256 workgroups

    AdaptiveMemory_wmma_kernel<<<grid, THREADS, 0, stream>>>(
        h_last, vids, mem, W_alpha, b_alpha, decay, out, B);
}